// TAM_20950850470193
// MI455X (gfx1250) — compile-verified
//
#include <hip/hip_runtime.h>
#include <math.h>

// ---------------------------------------------------------------------------
// TAM (Temporal Adaptive Module) for MI455X / gfx1250, wave32.
// Memory-bound: ~308 MB of traffic -> design target ~13us at 23.3 TB/s HBM.
// x (102.8 MB) fits in the 192 MB L2: K1 streams it in, K4 re-reads it from
// L2.  K4's output stores use non-temporal hints so the write stream does not
// evict x from L2.  The only matmul-shaped work (theta @ g_w1, 256x16x32 per
// batch) runs on V_WMMA_F32_16X16X4_F32.
// ---------------------------------------------------------------------------

typedef __attribute__((ext_vector_type(2))) float v2f;
typedef __attribute__((ext_vector_type(4))) float f4;
typedef __attribute__((ext_vector_type(8))) float v8f;

#define NB   8
#define TT   16
#define HH   28
#define WW   28
#define CC   256
#define CM   64
#define KK   3
#define AAF  32            // T*alpha features in global branch
#define HW   (HH*WW)       // 784
#define HWC  (HH*WW*CC)    // 200704
#define BN_EPS 1e-3f

// ---------------------------------------------------------------------------
// K0: zero theta accumulator (ws is poisoned with 0xAA by the harness).
// ---------------------------------------------------------------------------
__global__ void tam_zero(float* __restrict__ theta) {
    int i = blockIdx.x * blockDim.x + threadIdx.x;
    if (i < NB * CC * TT) theta[i] = 0.0f;
}

// ---------------------------------------------------------------------------
// K1: theta[b][c][t] = mean_{h,w} x[(b*T+t), h, w, c]
// grid = NB*TT*HH blocks of 256 threads (one channel per thread, coalesced).
// Default RT loads so x populates L2 for K4's re-read.
// ---------------------------------------------------------------------------
__global__ void tam_theta(const float* __restrict__ x, float* __restrict__ theta) {
    int row = blockIdx.x % HH;
    int bt  = blockIdx.x / HH;          // b*T + t
    int b   = bt / TT;
    int t   = bt % TT;
    int c   = threadIdx.x;

    const float* p = x + (size_t)bt * HWC + (size_t)row * WW * CC + c;
    float s = 0.0f;
#pragma unroll
    for (int w = 0; w < WW; ++w) s += p[w * CC];
    atomicAdd(&theta[(b * CC + c) * TT + t], s * (1.0f / (float)HW));
}

// ---------------------------------------------------------------------------
// K2: global branch.  Per batch b:
//   g = relu(BN(theta(256x16) @ g_w1(16x32)))   -- WMMA f32 16x16x4
//   kern = softmax(g @ g_w2(32x3))              -- per-channel scalar
// theta layout [b][c][t];  kern stored [b][k][c] for coalesced reads in K4.
// Block: 512 threads = 16 waves, wave `wid` owns channel rows [16*wid, 16*wid+16).
// ---------------------------------------------------------------------------
__global__ void tam_global(const float* __restrict__ theta,
                           const float* __restrict__ gw1,
                           const float* __restrict__ ggamma,
                           const float* __restrict__ gbeta,
                           const float* __restrict__ gmean,
                           const float* __restrict__ gvar,
                           const float* __restrict__ gw2,
                           float* __restrict__ kern) {
    __shared__ float gl[CC][AAF];       // 32 KB

    const int b    = blockIdx.x;
    const int tid  = threadIdx.x;
    const int lane = tid & 31;
    const int wid  = tid >> 5;
    const int m    = lane & 15;         // row (A) / col (B,C) index
    const int kh   = lane >> 4;         // lane-half selects K pair / M+8 rows
    const int c0   = wid * 16;

    const float* tb = theta + (size_t)b * CC * TT;

    v8f acc0 = {0.f,0.f,0.f,0.f,0.f,0.f,0.f,0.f};
    v8f acc1 = {0.f,0.f,0.f,0.f,0.f,0.f,0.f,0.f};

#if __has_builtin(__builtin_amdgcn_wmma_f32_16x16x4_f32)
#pragma unroll
    for (int ks = 0; ks < 4; ++ks) {
        const int k0 = ks * 4 + kh * 2;           // K offset for this lane half
        // A: 16x4 f32 tile of theta (M=channel, K=time)
        v2f A;
        A.x = tb[(c0 + m) * TT + k0];
        A.y = tb[(c0 + m) * TT + k0 + 1];
        // B: 4x16 f32 tiles of g_w1 (K=time, N=feature), N-tiles 0..15 / 16..31
        v2f B0, B1;
        B0.x = gw1[(k0)     * AAF + m];
        B0.y = gw1[(k0 + 1) * AAF + m];
        B1.x = gw1[(k0)     * AAF + 16 + m];
        B1.y = gw1[(k0 + 1) * AAF + 16 + m];
        acc0 = __builtin_amdgcn_wmma_f32_16x16x4_f32(false, A, false, B0,
                                                     (short)0, acc0, false, false);
        acc1 = __builtin_amdgcn_wmma_f32_16x16x4_f32(false, A, false, B1,
                                                     (short)0, acc1, false, false);
    }
#else
    // scalar fallback (should not be hit on gfx1250)
    for (int r = 0; r < 8; ++r) {
        float s0 = 0.f, s1 = 0.f;
        int ch = c0 + r + 8 * kh;
        for (int t = 0; t < TT; ++t) {
            s0 += tb[ch * TT + t] * gw1[t * AAF + m];
            s1 += tb[ch * TT + t] * gw1[t * AAF + 16 + m];
        }
        acc0[r] = s0; acc1[r] = s1;
    }
#endif

    // fused BN + ReLU on the accumulator, write g to LDS.
    // C/D layout: lane n holds column N=n, VGPR r holds row M=r (+8 for lanes>=16)
    const int a0 = m, a1 = 16 + m;
    const float sc0 = ggamma[a0] * rsqrtf(gvar[a0] + BN_EPS);
    const float sc1 = ggamma[a1] * rsqrtf(gvar[a1] + BN_EPS);
    const float mu0 = gmean[a0], mu1 = gmean[a1];
    const float be0 = gbeta[a0], be1 = gbeta[a1];
#pragma unroll
    for (int r = 0; r < 8; ++r) {
        const int ch = c0 + r + 8 * kh;
        float v0 = (acc0[r] - mu0) * sc0 + be0;
        float v1 = (acc1[r] - mu1) * sc1 + be1;
        gl[ch][a0] = v0 > 0.f ? v0 : 0.f;
        gl[ch][a1] = v1 > 0.f ? v1 : 0.f;
    }
    __syncthreads();

    // second matmul (32 -> 3) + softmax, one channel per thread
    if (tid < CC) {
        const int c = tid;
        float s0 = 0.f, s1 = 0.f, s2 = 0.f;
#pragma unroll
        for (int a = 0; a < AAF; ++a) {
            const float g = gl[c][a];
            s0 += g * gw2[a * KK + 0];
            s1 += g * gw2[a * KK + 1];
            s2 += g * gw2[a * KK + 2];
        }
        const float mx = fmaxf(s0, fmaxf(s1, s2));
        const float e0 = expf(s0 - mx), e1 = expf(s1 - mx), e2 = expf(s2 - mx);
        const float inv = 1.0f / (e0 + e1 + e2);
        kern[(b * KK + 0) * CC + c] = e0 * inv;
        kern[(b * KK + 1) * CC + c] = e1 * inv;
        kern[(b * KK + 2) * CC + c] = e2 * inv;
    }
}

// ---------------------------------------------------------------------------
// K3: local branch.  Per (b,t) block of 256 threads:
//   l1[m]  = relu(BN( sum_{dt,c} theta[b, t+dt-1, c] * l_w1[dt, c, m] ))
//   act[c] = sigmoid( sum_m l1[m] * l_w2[0, m, c] )
// act stored [b][t][c] for coalesced reads in K4.
// ---------------------------------------------------------------------------
__global__ void tam_local(const float* __restrict__ theta,
                          const float* __restrict__ lw1,
                          const float* __restrict__ lgamma,
                          const float* __restrict__ lbeta,
                          const float* __restrict__ lmean,
                          const float* __restrict__ lvar,
                          const float* __restrict__ lw2,
                          float* __restrict__ act) {
    __shared__ float th[3][CC];
    __shared__ float lb[CM];

    const int b = blockIdx.x / TT;
    const int t = blockIdx.x % TT;
    const int c = threadIdx.x;

#pragma unroll
    for (int dt = 0; dt < 3; ++dt) {
        const int tt = t + dt - 1;                      // SAME padding
        th[dt][c] = (tt >= 0 && tt < TT)
                  ? theta[((size_t)b * CC + c) * TT + tt] : 0.0f;
    }
    __syncthreads();

    if (c < CM) {
        float acc = 0.0f;
        for (int dt = 0; dt < 3; ++dt)
            for (int cc = 0; cc < CC; ++cc)
                acc += th[dt][cc] * lw1[(dt * CC + cc) * CM + c];
        float v = (acc - lmean[c]) * (lgamma[c] * rsqrtf(lvar[c] + BN_EPS)) + lbeta[c];
        lb[c] = v > 0.f ? v : 0.f;
    }
    __syncthreads();

    float acc2 = 0.0f;
#pragma unroll
    for (int mIdx = 0; mIdx < CM; ++mIdx)
        acc2 += lb[mIdx] * lw2[mIdx * CC + c];
    act[((size_t)b * TT + t) * CC + c] = 1.0f / (1.0f + expf(-acc2));
}

// ---------------------------------------------------------------------------
// K4: main bandwidth pass.
//   y[b,c,t,h,w] = k0*g[t-1] + k1*g[t] + k2*g[t+1],  g[t] = x[t]*act[b,c,t]
// One thread per (b, spatial, 4-channel group); b128 loads, rolling 3-tap
// window over T.  Output stores are NON-TEMPORAL: out is never re-read, and
// keeping the write stream out of L2 preserves x's residency there
// (x 102.8MB + out 102.8MB would exceed the 192MB L2 with RT stores).
// ---------------------------------------------------------------------------
__global__ void tam_main(const float* __restrict__ x,
                         const float* __restrict__ kern,
                         const float* __restrict__ act,
                         float* __restrict__ out) {
    const int gid  = blockIdx.x * blockDim.x + threadIdx.x;   // NB*HW*(CC/4)
    const int c4   = gid & 63;               // channel quad
    const int rest = gid >> 6;
    const int sp   = rest % HW;
    const int b    = rest / HW;
    if (b >= NB) return;

    const f4* x4 = (const f4*)x;
    const f4* k4 = (const f4*)kern;
    const f4* a4 = (const f4*)act;
    f4*       o4 = (f4*)out;

    const f4 k0 = k4[(b * KK + 0) * (CC / 4) + c4];
    const f4 k1 = k4[(b * KK + 1) * (CC / 4) + c4];
    const f4 k2 = k4[(b * KK + 2) * (CC / 4) + c4];

    const size_t step  = HWC / 4;
    const size_t base  = (size_t)(b * TT) * step + (size_t)sp * (CC / 4) + c4;
    const int    abase = (b * TT) * (CC / 4) + c4;

    f4 gprev = {0.f, 0.f, 0.f, 0.f};
    f4 gcur  = x4[base] * a4[abase];
#pragma unroll
    for (int t = 0; t < TT; ++t) {
        f4 gnext = {0.f, 0.f, 0.f, 0.f};
        if (t + 1 < TT)
            gnext = x4[base + (size_t)(t + 1) * step] * a4[abase + (t + 1) * (CC / 4)];
        f4 y = k0 * gprev + k1 * gcur + k2 * gnext;
        __builtin_nontemporal_store(y, &o4[base + (size_t)t * step]);
        gprev = gcur;
        gcur  = gnext;
    }
}

// ---------------------------------------------------------------------------
// Launch.  Inputs (all fp32, setup_inputs order):
//  0:x 1:g_w1 2:g_gamma 3:g_beta 4:g_mean 5:g_var 6:g_w2
//  7:l_w1 8:l_gamma 9:l_beta 10:l_mean 11:l_var 12:l_w2
// Workspace: theta (32768 f) | kern (6144 f) | act (32768 f)  = 280 KB.
// ---------------------------------------------------------------------------
extern "C" void kernel_launch(void* const* d_in, const int* in_sizes, int n_in,
                              void* d_out, int out_size, void* d_ws, size_t ws_size,
                              hipStream_t stream) {
    const float* x      = (const float*)d_in[0];
    const float* g_w1   = (const float*)d_in[1];
    const float* g_gam  = (const float*)d_in[2];
    const float* g_bet  = (const float*)d_in[3];
    const float* g_mean = (const float*)d_in[4];
    const float* g_var  = (const float*)d_in[5];
    const float* g_w2   = (const float*)d_in[6];
    const float* l_w1   = (const float*)d_in[7];
    const float* l_gam  = (const float*)d_in[8];
    const float* l_bet  = (const float*)d_in[9];
    const float* l_mean = (const float*)d_in[10];
    const float* l_var  = (const float*)d_in[11];
    const float* l_w2   = (const float*)d_in[12];
    float* out = (float*)d_out;

    float* theta = (float*)d_ws;            // NB*CC*TT = 32768 floats
    float* kern  = theta + NB * CC * TT;    // NB*KK*CC =  6144 floats
    float* act   = kern + NB * KK * CC;     // NB*TT*CC = 32768 floats

    tam_zero<<<(NB * CC * TT + 255) / 256, 256, 0, stream>>>(theta);
    tam_theta<<<NB * TT * HH, 256, 0, stream>>>(x, theta);
    tam_global<<<NB, 512, 0, stream>>>(theta, g_w1, g_gam, g_bet, g_mean, g_var,
                                       g_w2, kern);
    tam_local<<<NB * TT, 256, 0, stream>>>(theta, l_w1, l_gam, l_bet, l_mean,
                                           l_var, l_w2, act);
    tam_main<<<(NB * HW * (CC / 4)) / 256, 256, 0, stream>>>(x, kern, act, out);
}